// GCN_8770323219100
// MI455X (gfx1250) — compile-verified
//
#include <hip/hip_runtime.h>

typedef float v2f __attribute__((ext_vector_type(2)));
typedef float v8f __attribute__((ext_vector_type(8)));

#define HID 64
#define KMAX 128

// ---------------------------------------------------------------- elementwise
__global__ __launch_bounds__(256) void k_fill(float* p, float v, long long n) {
    long long i = (long long)blockIdx.x * blockDim.x + threadIdx.x;
    if (i < n) p[i] = v;
}

__global__ __launch_bounds__(256) void k_deg_edges(const long long* __restrict__ col,
                                                   float* __restrict__ deg, int E) {
    int e = blockIdx.x * blockDim.x + threadIdx.x;
    if (e < E) atomicAdd(&deg[(int)col[e]], 1.0f);
}

__global__ __launch_bounds__(256) void k_rsqrt_inplace(float* p, int n) {
    int i = blockIdx.x * blockDim.x + threadIdx.x;
    if (i < n) {
        float d = p[i];
        p[i] = d > 0.0f ? rsqrtf(d) : 0.0f;
    }
}

// xs[n][j] = dinv[n] * xl[n][j]
__global__ __launch_bounds__(256) void k_scale(const float* __restrict__ xl,
                                               const float* __restrict__ dinv,
                                               float* __restrict__ xs, long long total) {
    long long t = (long long)blockIdx.x * blockDim.x + threadIdx.x;
    if (t < total) xs[t] = dinv[t >> 6] * xl[t];
}

// h[n][j] = relu(dinv[n] * (agg[n][j] + dinv[n]*xl[n][j]) + b[j])   (in-place on xl OK)
__global__ __launch_bounds__(256) void k_combine(float* __restrict__ xl,
                                                 const float* __restrict__ agg,
                                                 const float* __restrict__ dinv,
                                                 const float* __restrict__ b,
                                                 long long total) {
    long long t = (long long)blockIdx.x * blockDim.x + threadIdx.x;
    if (t >= total) return;
    float di = dinv[t >> 6];
    float v = di * (agg[t] + di * xl[t]) + b[t & 63];
    xl[t] = v > 0.0f ? v : 0.0f;
}

// ---------------------------------------------------------------- edge scatter
// agg[c][j] += xs[r][j]; 64 consecutive threads share one edge (coalesced).
__global__ __launch_bounds__(256) void k_edge_scatter(const long long* __restrict__ rows,
                                                      const long long* __restrict__ cols,
                                                      const float* __restrict__ xs,
                                                      float* __restrict__ agg,
                                                      long long total) {
    long long t = (long long)blockIdx.x * blockDim.x + threadIdx.x;
    if (t >= total) return;
    long long e = t >> 6;
    int j = (int)(t & 63);
    int r = (int)rows[e];
    int c = (int)cols[e];
    atomicAdd(&agg[(long long)c * HID + j], xs[(long long)r * HID + j]);
}

// ---------------------------------------------------------------- WMMA GEMM
// out[N x 64] = A[N x K] * W[K x 64], f32 WMMA 16x16x4, one wave -> 16x64 slab.
// W staged in LDS, repacked so each lane's (K+kk, K+kk+1) pair at column n is one
// aligned b64 load hitting a unique bank pair (conflict-free across the wave):
//   wlds[(k>>2)*256 + n*4 + (k&3)] = W[k][n]
__global__ __launch_bounds__(256) void k_gemm_n64_wmma(const float* __restrict__ A,
                                                       const float* __restrict__ W,
                                                       float* __restrict__ out,
                                                       int nrows, int K) {
    __shared__ float wlds[KMAX * HID];

    // stage + repack W (K*64 floats, coalesced global reads)
    for (int t = threadIdx.x; t < K * HID; t += blockDim.x) {
        int k = t >> 6;
        int n = t & 63;
        wlds[(k & ~3) * HID + ((n << 2) | (k & 3))] = W[t];
    }
    __syncthreads();

    const int lane = threadIdx.x & 31;
    const int wave = threadIdx.x >> 5;
    const int tile = blockIdx.x * (blockDim.x >> 5) + wave;   // 16-row tile
    const int row0 = tile * 16;
    if (row0 >= nrows) return;

    const int m  = lane & 15;          // M (for A) / N (for B) index
    const int kk = (lane >> 4) * 2;    // K sub-offset: lanes 0-15 -> K+0,K+1; 16-31 -> K+2,K+3

    v8f acc0 = {}, acc1 = {}, acc2 = {}, acc3 = {};
    const float* arow = A + (long long)(row0 + m) * K;

    for (int k0 = 0; k0 < K; k0 += 4) {
        v2f a;
        a.x = arow[k0 + kk];
        a.y = arow[k0 + kk + 1];
        const float* lp = wlds + (k0 << 6) + kk;   // k0*64 + kk
        v2f b0 = *(const v2f*)(lp + ((0  + m) << 2));
        v2f b1 = *(const v2f*)(lp + ((16 + m) << 2));
        v2f b2 = *(const v2f*)(lp + ((32 + m) << 2));
        v2f b3 = *(const v2f*)(lp + ((48 + m) << 2));
        acc0 = __builtin_amdgcn_wmma_f32_16x16x4_f32(false, a, false, b0, (short)0, acc0, false, false);
        acc1 = __builtin_amdgcn_wmma_f32_16x16x4_f32(false, a, false, b1, (short)0, acc1, false, false);
        acc2 = __builtin_amdgcn_wmma_f32_16x16x4_f32(false, a, false, b2, (short)0, acc2, false, false);
        acc3 = __builtin_amdgcn_wmma_f32_16x16x4_f32(false, a, false, b3, (short)0, acc3, false, false);
    }

    // C/D layout: VGPR r, lanes 0-15 -> M=r, lanes 16-31 -> M=8+r; N = lane&15
    const int rbase = row0 + (lane >> 4) * 8;
    for (int r = 0; r < 8; ++r) {
        float* o = out + (long long)(rbase + r) * HID + m;
        o[0]  = acc0[r];
        o[16] = acc1[r];
        o[32] = acc2[r];
        o[48] = acc3[r];
    }
}

// ---------------------------------------------------------------- final layer
// out[n] = sigmoid(dot(h[n], Wfc) + bfc); one wave per node.
__global__ __launch_bounds__(256) void k_final(const float* __restrict__ h,
                                               const float* __restrict__ Wfc,
                                               const float* __restrict__ bfc,
                                               float* __restrict__ out, int n) {
    int gwave = (int)((blockIdx.x * (long long)blockDim.x + threadIdx.x) >> 5);
    int lane = threadIdx.x & 31;
    if (gwave >= n) return;
    const float* hr = h + (long long)gwave * HID;
    float s = hr[lane] * Wfc[lane] + hr[lane + 32] * Wfc[lane + 32];
    for (int off = 16; off > 0; off >>= 1) s += __shfl_xor(s, off, 32);
    if (lane == 0) out[gwave] = 1.0f / (1.0f + __expf(-(s + bfc[0])));
}

// ---------------------------------------------------------------- launch
extern "C" void kernel_launch(void* const* d_in, const int* in_sizes, int n_in,
                              void* d_out, int out_size, void* d_ws, size_t ws_size,
                              hipStream_t stream) {
    const float*     x   = (const float*)d_in[0];
    const long long* ei  = (const long long*)d_in[1];   // int64 per reference
    const float*     W1  = (const float*)d_in[2];
    const float*     b1  = (const float*)d_in[3];
    const float*     W2  = (const float*)d_in[4];
    const float*     b2  = (const float*)d_in[5];
    const float*     Wfc = (const float*)d_in[6];
    const float*     bfc = (const float*)d_in[7];
    float* out = (float*)d_out;

    const int D = 128;
    const int N = in_sizes[0] / D;
    const int E = in_sizes[1] / 2;
    const long long NH = (long long)N * HID;
    const long long EH = (long long)E * HID;

    const long long* rows = ei;
    const long long* cols = ei + E;

    // workspace layout
    float* ws = (float*)d_ws;
    long long nAlign = ((long long)N + 255) & ~255LL;
    float* dinv = ws;               // N floats (deg, then rsqrt in place)
    float* B1 = ws + nAlign;        // N*64
    float* B2 = B1 + NH;            // N*64
    float* B3 = B2 + NH;            // N*64

    const int T = 256;
    dim3 blk(T);
    int gN   = (N + T - 1) / T;
    int gNH  = (int)((NH + T - 1) / T);
    int gEH  = (int)((EH + T - 1) / T);
    int gE   = (E + T - 1) / T;
    int ntiles = (N + 15) / 16;
    int gGemm = (ntiles + 7) / 8;          // 8 waves/block
    int gFin  = (N + 7) / 8;               // 8 waves/block, 1 node/wave

    // degrees -> dinv
    k_fill<<<gN, blk, 0, stream>>>(dinv, 1.0f, N);      // self loop
    k_deg_edges<<<gE, blk, 0, stream>>>(cols, dinv, E);
    k_rsqrt_inplace<<<gN, blk, 0, stream>>>(dinv, N);

    // ---- layer 1
    k_gemm_n64_wmma<<<gGemm, blk, 0, stream>>>(x, W1, B1, N, D);       // xl1 = x@W1
    k_scale<<<gNH, blk, 0, stream>>>(B1, dinv, B2, NH);                // xs1
    k_fill<<<gNH, blk, 0, stream>>>(B3, 0.0f, NH);                     // agg1 = 0
    k_edge_scatter<<<gEH, blk, 0, stream>>>(rows, cols, B2, B3, EH);
    k_combine<<<gNH, blk, 0, stream>>>(B1, B3, dinv, b1, NH);          // h1 (in B1)

    // ---- layer 2
    k_gemm_n64_wmma<<<gGemm, blk, 0, stream>>>(B1, W2, B2, N, HID);    // xl2 = h1@W2
    k_scale<<<gNH, blk, 0, stream>>>(B2, dinv, B3, NH);                // xs2
    k_fill<<<gNH, blk, 0, stream>>>(B1, 0.0f, NH);                     // agg2 = 0
    k_edge_scatter<<<gEH, blk, 0, stream>>>(rows, cols, B3, B1, EH);
    k_combine<<<gNH, blk, 0, stream>>>(B2, B1, dinv, b2, NH);          // h2 (in B2)

    // ---- readout
    k_final<<<gFin, blk, 0, stream>>>(B2, Wfc, bfc, out, N);
}